// GraphAttentionLayer_16054587752770
// MI455X (gfx1250) — compile-verified
//
#include <hip/hip_runtime.h>
#include <hip/hip_bf16.h>

typedef float v2f __attribute__((ext_vector_type(2)));
typedef float v8f __attribute__((ext_vector_type(8)));

#define NTOK   16384      // B*N tokens
#define DMODEL 128
#define NGRAPH 1024
#define ALPHA  0.2f
#define BN_EPS 1e-5f

// ---------------------------------------------------------------------------
// Kernel 1: Wh = h @ W^T via V_WMMA_F32_16X16X4_F32.
// Block = 256 threads (8 waves). Each wave: one 16-row M tile x all 8 N tiles.
// W (128x128 f32, 64KB) staged in LDS once per block.
// ---------------------------------------------------------------------------
__global__ __launch_bounds__(256)
void gat_gemm_wmma(const float* __restrict__ h, const float* __restrict__ W,
                   float* __restrict__ Wh) {
    __shared__ float Wlds[DMODEL * DMODEL];   // 64 KB

    const int tid = threadIdx.x;
    // cooperative W load: 16384 floats / 256 threads = 16 float4 each
    const float4* Wg = (const float4*)W;
    float4* Wl = (float4*)Wlds;
#pragma unroll
    for (int it = 0; it < 16; ++it)
        Wl[tid + it * 256] = Wg[tid + it * 256];
    __syncthreads();

    const int wave  = tid >> 5;
    const int lane  = tid & 31;
    const int mtile = blockIdx.x * 8 + wave;          // 0..1023
    const int mrow  = (lane & 15);                    // row-in-tile for A / col for B
    const int khalf = (lane >> 4) << 1;               // 0 or 2 : K pair base

    v8f acc[8];
#pragma unroll
    for (int t = 0; t < 8; ++t) acc[t] = (v8f){0.f,0.f,0.f,0.f,0.f,0.f,0.f,0.f};

    const float* hrow = h + (mtile * 16 + mrow) * DMODEL + khalf;

    for (int kk = 0; kk < DMODEL; kk += 4) {
        // A frag: A[m, kk+khalf .. +1]  (contiguous float2 in h)
        v2f a = *(const v2f*)(hrow + kk);
#pragma unroll
        for (int t = 0; t < 8; ++t) {
            // B frag: B[k][n] = W[n][k]; lane holds n = mrow of N-tile t,
            // k pair = kk+khalf.. contiguous along W's row.
            v2f b = *(const v2f*)(&Wlds[(t * 16 + mrow) * DMODEL + kk + khalf]);
            acc[t] = __builtin_amdgcn_wmma_f32_16x16x4_f32(
                false, a, false, b, (short)0, acc[t], false, false);
        }
    }

    // C/D layout: lane l, vgpr v -> row M = v + 8*(l>=16), col N = l&15
    const int hi = lane >> 4;
    const int cn = lane & 15;
#pragma unroll
    for (int t = 0; t < 8; ++t) {
#pragma unroll
        for (int v = 0; v < 8; ++v) {
            int r = mtile * 16 + v + 8 * hi;
            Wh[r * DMODEL + t * 16 + cn] = acc[t][v];
        }
    }
}

// ---------------------------------------------------------------------------
// Kernel 2: s1[r] = Wh[r,:].a1 ; s2[r] = Wh[r,:].a2   (one wave per row)
// ---------------------------------------------------------------------------
__global__ __launch_bounds__(256)
void gat_scores(const float* __restrict__ Wh, const float* __restrict__ a1,
                const float* __restrict__ a2, float* __restrict__ s1,
                float* __restrict__ s2) {
    const int wave = threadIdx.x >> 5;
    const int lane = threadIdx.x & 31;
    const int r = blockIdx.x * 8 + wave;

    const float4 x = ((const float4*)(Wh + r * DMODEL))[lane];
    const float4 u = ((const float4*)a1)[lane];
    const float4 v = ((const float4*)a2)[lane];
    float d1 = x.x * u.x + x.y * u.y + x.z * u.z + x.w * u.w;
    float d2 = x.x * v.x + x.y * v.y + x.z * v.z + x.w * v.w;
#pragma unroll
    for (int off = 16; off > 0; off >>= 1) {
        d1 += __shfl_xor(d1, off, 32);
        d2 += __shfl_xor(d2, off, 32);
    }
    if (lane == 0) { s1[r] = d1; s2[r] = d2; }
}

// ---------------------------------------------------------------------------
// Kernel 3: tridiagonal softmax + stencil combine + ELU; emit per-block
// deterministic partial sums for BatchNorm.
// Block handles 128 rows; thread (d = t&127, half = t>>7) covers 64 rows.
// ---------------------------------------------------------------------------
__global__ __launch_bounds__(256)
void gat_attn(const float* __restrict__ Wh, const float* __restrict__ s1,
              const float* __restrict__ s2, const float* __restrict__ abp,
              float* __restrict__ out, float* __restrict__ psum,
              float* __restrict__ psq) {
    const float ab = abp[0];
    const int d    = threadIdx.x & 127;
    const int half = threadIdx.x >> 7;
    const int r0   = blockIdx.x * 128;

    float sum = 0.f, sq = 0.f;
    for (int rl = half * 64; rl < half * 64 + 64; ++rl) {
        const int g = r0 + rl;
        const int i = g & (NGRAPH - 1);
        const float sc = s1[g];
        const bool hasl = (i > 0), hasr = (i < NGRAPH - 1);

        float ec = sc + s2[g] + ab;        ec = fmaxf(ec, ALPHA * ec);
        float el = -3.0e38f, er = -3.0e38f;
        if (hasl) { el = sc + s2[g - 1] + ab; el = fmaxf(el, ALPHA * el); }
        if (hasr) { er = sc + s2[g + 1] + ab; er = fmaxf(er, ALPHA * er); }

        const float m  = fmaxf(ec, fmaxf(el, er));
        const float wl = hasl ? __expf(el - m) : 0.f;
        const float wc = __expf(ec - m);
        const float wr = hasr ? __expf(er - m) : 0.f;
        const float inv = 1.f / (wl + wc + wr);

        float x = wc * Wh[g * DMODEL + d];
        if (hasl) x += wl * Wh[(g - 1) * DMODEL + d];
        if (hasr) x += wr * Wh[(g + 1) * DMODEL + d];
        x *= inv;
        x = (x > 0.f) ? x : (__expf(x) - 1.f);      // ELU

        out[g * DMODEL + d] = x;
        sum += x; sq += x * x;
    }

    __shared__ float ls[128], lq[128];
    if (half == 1) { ls[d] = sum; lq[d] = sq; }
    __syncthreads();
    if (half == 0) {
        psum[blockIdx.x * 128 + d] = sum + ls[d];
        psq [blockIdx.x * 128 + d] = sq  + lq[d];
    }
}

// ---------------------------------------------------------------------------
// Kernel 4: fold partials -> per-feature scale/shift (1 block, fixed order)
// ---------------------------------------------------------------------------
__global__ void gat_stats(const float* __restrict__ psum, const float* __restrict__ psq,
                          const float* __restrict__ gamma, const float* __restrict__ beta,
                          float* __restrict__ scale, float* __restrict__ shift) {
    const int d = threadIdx.x;   // 128 threads
    float s = 0.f, q = 0.f;
    for (int b = 0; b < 128; ++b) { s += psum[b * 128 + d]; q += psq[b * 128 + d]; }
    const float mean = s * (1.f / (float)NTOK);
    const float var  = q * (1.f / (float)NTOK) - mean * mean;
    const float invs = rsqrtf(var + BN_EPS);
    const float sc   = gamma[d] * invs;
    scale[d] = sc;
    shift[d] = beta[d] - mean * sc;
}

// ---------------------------------------------------------------------------
// Kernel 5: apply BN in place on d_out (float4)
// ---------------------------------------------------------------------------
__global__ __launch_bounds__(256)
void gat_bn(float* __restrict__ out, const float* __restrict__ scale,
            const float* __restrict__ shift) {
    const int idx = blockIdx.x * 256 + threadIdx.x;   // float4 index
    float4* o = (float4*)out;
    float4 x = o[idx];
    const int d4 = (idx & 31) * 4;                    // 32 float4 per row
    x.x = x.x * scale[d4 + 0] + shift[d4 + 0];
    x.y = x.y * scale[d4 + 1] + shift[d4 + 1];
    x.z = x.z * scale[d4 + 2] + shift[d4 + 2];
    x.w = x.w * scale[d4 + 3] + shift[d4 + 3];
    o[idx] = x;
}

// ---------------------------------------------------------------------------
extern "C" void kernel_launch(void* const* d_in, const int* in_sizes, int n_in,
                              void* d_out, int out_size, void* d_ws, size_t ws_size,
                              hipStream_t stream) {
    const float* h     = (const float*)d_in[0];
    const float* W     = (const float*)d_in[1];
    const float* a1    = (const float*)d_in[2];
    const float* a2    = (const float*)d_in[3];
    const float* ab    = (const float*)d_in[4];
    const float* gamma = (const float*)d_in[5];
    const float* beta  = (const float*)d_in[6];
    float* out = (float*)d_out;

    float* ws    = (float*)d_ws;
    float* Wh    = ws;                         // 16384*128
    float* s1    = ws + 2097152;               // 16384
    float* s2    = s1 + NTOK;                  // 16384
    float* psum  = s2 + NTOK;                  // 128*128
    float* psq   = psum + NTOK;                // 128*128
    float* scale = psq + NTOK;                 // 128
    float* shift = scale + 128;                // 128

    gat_gemm_wmma<<<128, 256, 0, stream>>>(h, W, Wh);
    gat_scores  <<<2048, 256, 0, stream>>>(Wh, a1, a2, s1, s2);
    gat_attn    <<<128, 256, 0, stream>>>(Wh, s1, s2, ab, out, psum, psq);
    gat_stats   <<<1, 128, 0, stream>>>(psum, psq, gamma, beta, scale, shift);
    gat_bn      <<<2048, 256, 0, stream>>>(out, scale, shift);
}